// RKDLoss_9242769621708
// MI455X (gfx1250) — compile-verified
//
#include <hip/hip_runtime.h>
#include <hip/hip_bf16.h>

// RKD loss via Gram-matrix factorization.
//   G = E E^T (WMMA f32 16x16x4, exact fp32)
//   dist[i,j]  = sqrt(clip(Gii+Gjj-2Gij, EPS)) off-diag
//   angle[i,j,k] = h_ij h_ik (G_jk - G_ij - G_ik + G_ii),  h_ij = 1/max(r_ij, EPS)
// All reductions are deterministic (block partials + fixed-order fold).

typedef __attribute__((ext_vector_type(2))) float v2f;
typedef __attribute__((ext_vector_type(8))) float v8f;

#define RKD_EPS 1e-12f

// ---------------- block reduction (deterministic within block) -------------
__device__ __forceinline__ float block_reduce_sum(float v) {
    __shared__ float sm[8];
    __syncthreads();                       // protect sm reuse across calls
    int lane = threadIdx.x & 31;
    int wv   = threadIdx.x >> 5;
    #pragma unroll
    for (int o = 16; o > 0; o >>= 1) v += __shfl_down(v, o, 32);
    if (lane == 0) sm[wv] = v;
    __syncthreads();
    if (wv == 0) {
        v = (lane < (int)(blockDim.x >> 5)) ? sm[lane] : 0.0f;
        #pragma unroll
        for (int o = 4; o > 0; o >>= 1) v += __shfl_down(v, o, 32);
    }
    return v;  // valid in thread 0
}

// ---------------- Gram: G = E E^T via v_wmma_f32_16x16x4_f32 ---------------
// 512 wave-tiles total: 2 inputs x (16x16 grid of 16x16 tiles). 8 waves/block.
__global__ __launch_bounds__(256) void rkd_gram_kernel(const float* __restrict__ fs,
                                                       const float* __restrict__ ft,
                                                       float* __restrict__ Gs,
                                                       float* __restrict__ Gt) {
    const int D = 1024;
    int wave  = threadIdx.x >> 5;
    int lane  = threadIdx.x & 31;
    int tile  = blockIdx.x * 8 + wave;     // 0..511
    int which = tile >> 8;                 // 0: student, 1: teacher
    int t     = tile & 255;
    int r0    = (t >> 4) << 4;
    int c0    = (t & 15) << 4;
    const float* E = which ? ft : fs;
    float*       G = which ? Gt : Gs;

    int half = lane >> 4;                  // 0 or 1
    int l    = lane & 15;

    const float* arow = E + (size_t)(r0 + l) * D;
    const float* brow = E + (size_t)(c0 + l) * D;

    v8f acc = {0.f,0.f,0.f,0.f,0.f,0.f,0.f,0.f};

    for (int k0 = 0; k0 < D; k0 += 4) {
        // A 16x4: lanes 0-15 hold K=k0,k0+1 ; lanes 16-31 hold K=k0+2,k0+3
        float4 av = *(const float4*)(arow + k0);
        v2f a;
        a.x = half ? av.z : av.x;
        a.y = half ? av.w : av.y;
        // B 4x16 (= E rows of the column block): VGPR0 lanes0-15 K=k0,
        // lanes16-31 K=k0+1 ; VGPR1 lanes0-15 K=k0+2, lanes16-31 K=k0+3
        float4 bv = *(const float4*)(brow + k0);
        v2f b;
        b.x = half ? bv.y : bv.x;
        b.y = half ? bv.w : bv.z;
        acc = __builtin_amdgcn_wmma_f32_16x16x4_f32(
            /*neg_a=*/false, a, /*neg_b=*/false, b,
            /*c_mod=*/(short)0, acc, /*reuse_a=*/false, /*reuse_b=*/false);
    }
    // C/D layout: VGPR v -> row r0 + v + half*8, col c0 + l
    #pragma unroll
    for (int v = 0; v < 8; v++) {
        G[(size_t)(r0 + v + half * 8) * 256 + (c0 + l)] = acc[v];
    }
}

// -------- per-pair prep: pdist D, inverse norms H, masked-mean partials ----
__global__ __launch_bounds__(256) void rkd_prep_kernel(const float* __restrict__ G,
                                                       float* __restrict__ Dm,
                                                       float* __restrict__ Hm,
                                                       float* __restrict__ Psum,
                                                       float* __restrict__ Pcnt) {
    int i   = blockIdx.x;          // row
    int j   = threadIdx.x;         // col
    int idx = i * 256 + j;
    float gii = G[i * 257];
    float gjj = G[j * 257];
    float gij = G[idx];
    float raw = gii + gjj - 2.0f * gij;
    float d   = (i == j) ? 0.0f : sqrtf(fmaxf(raw, RKD_EPS));
    Dm[idx] = d;
    Hm[idx] = 1.0f / fmaxf(sqrtf(fmaxf(raw, 0.0f)), RKD_EPS);
    float m = (d > 0.0f) ? 1.0f : 0.0f;
    float s1 = block_reduce_sum(d);      // d*mask == d (d >= 0)
    float s2 = block_reduce_sum(m);
    if (threadIdx.x == 0) { Psum[i] = s1; Pcnt[i] = s2; }
}

// -------- fold masked-mean partials into mu_s, mu_t (deterministic) --------
__global__ void rkd_stats_kernel(const float* __restrict__ Psum_s,
                                 const float* __restrict__ Pcnt_s,
                                 const float* __restrict__ Psum_t,
                                 const float* __restrict__ Pcnt_t,
                                 float* __restrict__ mus) {
    float ss = 0.f, cs = 0.f, st = 0.f, ct = 0.f;
    for (int k = 0; k < 256; k++) {
        ss += Psum_s[k]; cs += Pcnt_s[k];
        st += Psum_t[k]; ct += Pcnt_t[k];
    }
    mus[0] = ss / cs;
    mus[1] = st / ct;
}

// -------- distance-branch smooth-L1 partials -------------------------------
__global__ __launch_bounds__(256) void rkd_dist_kernel(const float* __restrict__ Ds,
                                                       const float* __restrict__ Dt,
                                                       const float* __restrict__ mus,
                                                       float* __restrict__ Pdist) {
    float inv_mu_s = 1.0f / mus[0];
    float inv_mu_t = 1.0f / mus[1];
    int idx = blockIdx.x * 256 + threadIdx.x;
    float diff = fabsf(Ds[idx] * inv_mu_s - Dt[idx] * inv_mu_t);
    float v = (diff < 1.0f) ? 0.5f * diff * diff : diff - 0.5f;
    float tot = block_reduce_sum(v);
    if (threadIdx.x == 0) Pdist[blockIdx.x] = tot;
}

// -------- angle-branch smooth-L1 partials (one block per anchor i) ---------
__global__ __launch_bounds__(256) void rkd_angle_kernel(const float* __restrict__ Gs,
                                                        const float* __restrict__ Gt,
                                                        const float* __restrict__ Hs,
                                                        const float* __restrict__ Ht,
                                                        float* __restrict__ Pangle) {
    int i = blockIdx.x;
    int t = threadIdx.x;       // k index
    __shared__ float ps[256], qs[256], pt[256], qt[256];
    ps[t] = Hs[i * 256 + t];
    qs[t] = Gs[i * 256 + t];
    pt[t] = Ht[i * 256 + t];
    qt[t] = Gt[i * 256 + t];
    __syncthreads();
    float gii_s = qs[i];
    float gii_t = qt[i];
    float pks = ps[t], qks = qs[t];
    float pkt = pt[t], qkt = qt[t];
    float sum = 0.0f;
    for (int j = 0; j < 256; j++) {
        float gjs = Gs[j * 256 + t];
        float gjt = Gt[j * 256 + t];
        float as  = ps[j] * pks * (gjs - qs[j] - qks + gii_s);
        float at  = pt[j] * pkt * (gjt - qt[j] - qkt + gii_t);
        float diff = fabsf(as - at);
        sum += (diff < 1.0f) ? 0.5f * diff * diff : diff - 0.5f;
    }
    float tot = block_reduce_sum(sum);
    if (threadIdx.x == 0) Pangle[i] = tot;
}

// -------- final fold: loss = W_D*mean(sl1_d) + W_A*mean(sl1_a) -------------
__global__ void rkd_final_kernel(const float* __restrict__ Pdist,
                                 const float* __restrict__ Pangle,
                                 float* __restrict__ out) {
    float sd = 0.f, sa = 0.f;
    for (int k = 0; k < 256; k++) { sd += Pdist[k]; sa += Pangle[k]; }
    out[0] = 25.0f * (sd / 65536.0f) + 50.0f * (sa / 16777216.0f);
}

extern "C" void kernel_launch(void* const* d_in, const int* in_sizes, int n_in,
                              void* d_out, int out_size, void* d_ws, size_t ws_size,
                              hipStream_t stream) {
    (void)in_sizes; (void)n_in; (void)out_size; (void)ws_size;
    const float* fs = (const float*)d_in[0];
    const float* ft = (const float*)d_in[1];
    float* ws = (float*)d_ws;

    float* Gs     = ws;                // 65536
    float* Gt     = ws + 65536;        // 65536
    float* Hs     = ws + 131072;       // 65536
    float* Ht     = ws + 196608;       // 65536
    float* Ds     = ws + 262144;       // 65536
    float* Dt     = ws + 327680;       // 65536
    float* Psum_s = ws + 393216;       // 256
    float* Pcnt_s = ws + 393472;       // 256
    float* Psum_t = ws + 393728;       // 256
    float* Pcnt_t = ws + 393984;       // 256
    float* Pdist  = ws + 394240;       // 256
    float* Pangle = ws + 394496;       // 256
    float* mus    = ws + 394752;       // 2

    rkd_gram_kernel<<<64, 256, 0, stream>>>(fs, ft, Gs, Gt);
    rkd_prep_kernel<<<256, 256, 0, stream>>>(Gs, Ds, Hs, Psum_s, Pcnt_s);
    rkd_prep_kernel<<<256, 256, 0, stream>>>(Gt, Dt, Ht, Psum_t, Pcnt_t);
    rkd_stats_kernel<<<1, 1, 0, stream>>>(Psum_s, Pcnt_s, Psum_t, Pcnt_t, mus);
    rkd_dist_kernel<<<256, 256, 0, stream>>>(Ds, Dt, mus, Pdist);
    rkd_angle_kernel<<<256, 256, 0, stream>>>(Gs, Gt, Hs, Ht, Pangle);
    rkd_final_kernel<<<1, 1, 0, stream>>>(Pdist, Pangle, (float*)d_out);
}